// RQSCouplingLayer_45114336477673
// MI455X (gfx1250) — compile-verified
//
#include <hip/hip_runtime.h>

// RQS coupling layer forward: z = spline(x), log_jac.
// N = 16,777,216 f32, K = 5 bins, tail_bound = 2.5.
// HBM-bound: 12 B/elt -> ~201 MB -> ~8.6 us floor @ 23.3 TB/s.
// All bin-uniform math (softmax/cumsum/softplus/log s) hoisted to a
// per-block LDS table; hot loop is b128 vector IO + ds_load_b128 gathers.

typedef __attribute__((ext_vector_type(4))) float fvec4;

#define TAIL 2.5f
#define KBINS 5

__device__ __forceinline__ float softplusf(float v) {
  // log(1 + e^v), overflow-safe
  return fmaxf(v, 0.0f) + log1pf(expf(-fabsf(v)));
}

__device__ __forceinline__ void rqs_one(float x,
                                        const float* __restrict__ s_tbl,
                                        const float* __restrict__ s_bounds,
                                        float& z, float& lj) {
  const bool inside = fabsf(x) <= TAIL;
  const float xs = inside ? x : 0.0f;

  // searchsorted(cum_w[1:-1], xs, side='left') == count of boundaries < xs
  int idx = 0;
  idx += (s_bounds[0] < xs) ? 1 : 0;
  idx += (s_bounds[1] < xs) ? 1 : 0;
  idx += (s_bounds[2] < xs) ? 1 : 0;
  idx += (s_bounds[3] < xs) ? 1 : 0;

  // per-bin constants: A = {x_k, 1/(dx+eps), y_k, dy}
  //                    B = {d_k, d_k1, s_k, d_k+d_k1-2*s_k}
  //                    C =  2*log(s_k+eps)
  const fvec4 A = *reinterpret_cast<const fvec4*>(s_tbl + idx * 16);
  const fvec4 B = *reinterpret_cast<const fvec4*>(s_tbl + idx * 16 + 4);
  const float log2s = s_tbl[idx * 16 + 8];

  float xi = (xs - A.x) * A.y;
  xi = fminf(fmaxf(xi, 0.0f), 1.0f);
  const float omx   = 1.0f - xi;
  const float xi2   = xi * xi;
  const float xiomx = xi * omx;

  const float denom = fmaf(B.w, xiomx, B.z);                 // s + (d_k+d_k1-2s)*xi*(1-xi)
  const float num   = A.w * fmaf(B.z, xi2, B.x * xiomx);     // dy*(s*xi^2 + d_k*xi*(1-xi))
  const float zin   = A.z + __fdividef(num, denom + 1e-8f);

  const float arg = fmaf(B.y, xi2,
                    fmaf(2.0f * B.z, xiomx, B.x * (omx * omx))) + 1e-8f;
  const float ljin = log2s + __logf(arg) - 2.0f * __logf(fabsf(denom) + 1e-8f);

  z  = inside ? zin  : x;
  lj = inside ? ljin : 0.0f;
}

__global__ __launch_bounds__(256) void rqs_kernel(
    const float* __restrict__ x,
    const float* __restrict__ params,
    float* __restrict__ zout,
    float* __restrict__ ljout,
    int n, int n4) {
  __shared__ __align__(16) float s_tbl[KBINS * 16];
  __shared__ float s_bounds[4];

  if (threadIdx.x == 0) {
    float W[KBINS], H[KBINS], D[KBINS + 1];
    float cw[KBINS + 1], ch[KBINS + 1];

    for (int i = 0; i < KBINS; ++i) { W[i] = params[i]; H[i] = params[KBINS + i]; }
    for (int i = 0; i < KBINS + 1; ++i) D[i] = softplusf(params[2 * KBINS + i]) + 1e-5f;

    float mw = W[0], mh = H[0];
    for (int i = 1; i < KBINS; ++i) { mw = fmaxf(mw, W[i]); mh = fmaxf(mh, H[i]); }
    float sw = 0.0f, sh = 0.0f;
    for (int i = 0; i < KBINS; ++i) {
      W[i] = expf(W[i] - mw); sw += W[i];
      H[i] = expf(H[i] - mh); sh += H[i];
    }
    const float fw = (2.0f * TAIL) / sw;
    const float fh = (2.0f * TAIL) / sh;
    cw[0] = -TAIL; ch[0] = -TAIL;
    for (int i = 0; i < KBINS; ++i) {
      W[i] *= fw; H[i] *= fh;
      cw[i + 1] = cw[i] + W[i];
      ch[i + 1] = ch[i] + H[i];
    }
    for (int k = 0; k < KBINS; ++k) {
      const float s = H[k] / W[k];
      float* t = s_tbl + k * 16;
      t[0] = cw[k];
      t[1] = 1.0f / (cw[k + 1] - cw[k] + 1e-8f);
      t[2] = ch[k];
      t[3] = ch[k + 1] - ch[k];
      t[4] = D[k];
      t[5] = D[k + 1];
      t[6] = s;
      t[7] = D[k] + D[k + 1] - 2.0f * s;
      t[8] = 2.0f * logf(s + 1e-8f);
      t[9] = 0.0f; t[10] = 0.0f; t[11] = 0.0f;
      t[12] = 0.0f; t[13] = 0.0f; t[14] = 0.0f; t[15] = 0.0f;
    }
    s_bounds[0] = cw[1]; s_bounds[1] = cw[2];
    s_bounds[2] = cw[3]; s_bounds[3] = cw[4];
  }
  __syncthreads();

  const int stride = (int)(gridDim.x * blockDim.x);
  const int gid = (int)(blockIdx.x * blockDim.x + threadIdx.x);

  const fvec4* __restrict__ x4 = reinterpret_cast<const fvec4*>(x);
  fvec4* __restrict__ z4 = reinterpret_cast<fvec4*>(zout);
  fvec4* __restrict__ l4 = reinterpret_cast<fvec4*>(ljout);

  for (int i = gid; i < n4; i += stride) {
    __builtin_prefetch(&x4[i + stride], 0, 0);       // global_prefetch_b8
    const fvec4 xv = __builtin_nontemporal_load(&x4[i]);
    float z0, z1, z2, z3, l0, l1, l2, l3;
    rqs_one(xv.x, s_tbl, s_bounds, z0, l0);
    rqs_one(xv.y, s_tbl, s_bounds, z1, l1);
    rqs_one(xv.z, s_tbl, s_bounds, z2, l2);
    rqs_one(xv.w, s_tbl, s_bounds, z3, l3);
    const fvec4 zv = {z0, z1, z2, z3};
    const fvec4 lv = {l0, l1, l2, l3};
    __builtin_nontemporal_store(zv, &z4[i]);          // single-pass stream: NT
    __builtin_nontemporal_store(lv, &l4[i]);
  }

  // tail (n not divisible by 4; N = 2^24 so normally empty)
  for (int i = n4 * 4 + gid; i < n; i += stride) {
    float zz, ll;
    rqs_one(x[i], s_tbl, s_bounds, zz, ll);
    zout[i] = zz;
    ljout[i] = ll;
  }
}

extern "C" void kernel_launch(void* const* d_in, const int* in_sizes, int n_in,
                              void* d_out, int out_size, void* d_ws, size_t ws_size,
                              hipStream_t stream) {
  (void)n_in; (void)out_size; (void)d_ws; (void)ws_size;
  const float* x = (const float*)d_in[0];
  const float* params = (const float*)d_in[1];
  const int n = in_sizes[0];
  float* zout = (float*)d_out;
  float* ljout = zout + n;
  const int n4 = n >> 2;

  int grid = 4096;                       // grid-stride: 4 fvec4 iters/thread at N=2^24
  int maxg = n4 > 0 ? (n4 + 255) / 256 : 1;
  if (grid > maxg) grid = maxg;
  if (grid < 1) grid = 1;

  rqs_kernel<<<dim3(grid), dim3(256), 0, stream>>>(x, params, zout, ljout, n, n4);
}